// SelfTrainingLoss_24885040513252
// MI455X (gfx1250) — compile-verified
//
#include <hip/hip_runtime.h>
#include <math.h>

typedef float v2f __attribute__((ext_vector_type(2)));
typedef float v8f __attribute__((ext_vector_type(8)));

namespace {
constexpr int  kB   = 8;
constexpr int  kC   = 19;
constexpr int  kN   = 512 * 1024;            // pixels per image (2^19)
constexpr long kP   = (long)kB * kN;         // total pixels
constexpr int  kBC  = kB * kC;               // 152
constexpr float kConf = 0.9f;
constexpr float kFrac = 0.66f;

// workspace layout (bytes)
constexpr size_t OFF_SUM    = 0;
constexpr size_t OFF_CNT    = 4;
constexpr size_t OFF_COUNTS = 8;                       // 152 int
constexpr size_t OFF_KCUR   = 616;                     // 152 int
constexpr size_t OFF_PATH   = 1224;                    // 152 u32
constexpr size_t OFF_THR    = 1832;                    // 152 u32
constexpr size_t OFF_H1     = 2560;
constexpr size_t HBYTES     = (size_t)kBC * 256 * 4;   // 155648
constexpr size_t OFF_H2     = OFF_H1 + HBYTES;
constexpr size_t OFF_H3     = OFF_H2 + HBYTES;
constexpr size_t OFF_H4     = OFF_H3 + HBYTES;
constexpr size_t OFF_PROB   = OFF_H4 + HBYTES;         // 625152
constexpr size_t OFF_LAB    = OFF_PROB + (size_t)kP * 4;
constexpr int    ZERO_INTS  = (int)(OFF_PROB / 4);     // zero everything before prob
}

__global__ void zero_ws_kernel(int* __restrict__ w, int n) {
  int i = blockIdx.x * blockDim.x + threadIdx.x;
  if (i < n) w[i] = 0;
}

// Pass 1: softmax max-prob + argmax per pixel; per-class counts (LDS reduced);
// level-1 radix histogram over float-bit pattern (bits 30:23).
__global__ __launch_bounds__(256) void pass1_kernel(
    const float* __restrict__ pred, float* __restrict__ prob,
    unsigned char* __restrict__ lab, int* __restrict__ counts,
    int* __restrict__ h1) {
  __shared__ int scnt[kC];
  if (threadIdx.x < kC) scnt[threadIdx.x] = 0;
  __syncthreads();

  long t = (long)blockIdx.x * blockDim.x + threadIdx.x;
  long g = t * 4;                         // first pixel of this thread
  int  b = (int)(g >> 19);                // g / kN
  long n = g & (kN - 1);
  const float* base = pred + (size_t)b * kC * kN + n;

  float xs[kC * 4];
#pragma unroll
  for (int c = 0; c < kC; ++c) {
    float4 v = *(const float4*)(base + (size_t)c * kN);
    xs[c * 4 + 0] = v.x; xs[c * 4 + 1] = v.y;
    xs[c * 4 + 2] = v.z; xs[c * 4 + 3] = v.w;
  }

  float pv[4]; int av[4];
#pragma unroll
  for (int j = 0; j < 4; ++j) {
    float m = xs[j]; int a = 0;
#pragma unroll
    for (int c = 1; c < kC; ++c) {
      float x = xs[c * 4 + j];
      if (x > m) { m = x; a = c; }
    }
    float s = 0.f;
#pragma unroll
    for (int c = 0; c < kC; ++c) s += __expf(xs[c * 4 + j] - m);
    pv[j] = 1.0f / s;                     // max softmax prob
    av[j] = a;
    atomicAdd(&scnt[a], 1);
    unsigned vb = __float_as_uint(pv[j]);
    atomicAdd(&h1[(b * kC + a) * 256 + (int)(vb >> 23)], 1);
  }
  *(float4*)(prob + g) = make_float4(pv[0], pv[1], pv[2], pv[3]);
  *(uchar4*)(lab + g)  = make_uchar4((unsigned char)av[0], (unsigned char)av[1],
                                     (unsigned char)av[2], (unsigned char)av[3]);
  __syncthreads();
  if (threadIdx.x < kC) atomicAdd(&counts[b * kC + threadIdx.x], scnt[threadIdx.x]);
}

// Radix-select one 8-bit level: one wave per (image,class).
// Descending cumulative sum over 256 bins computed as triangular matmul:
// P = L(16x16, lower-tri ones) @ X(16x16, cols = 16-bin chunks) via
// 4 chained V_WMMA_F32_16X16X4_F32, then a shuffle scan over chunk totals.
template <int LEVEL>
__global__ __launch_bounds__(32) void select_kernel(
    const int* __restrict__ hist, const int* __restrict__ counts,
    int* __restrict__ kcur, unsigned* __restrict__ path,
    unsigned* __restrict__ thr) {
  const int idx  = blockIdx.x;            // b*19 + c
  const int lane = threadIdx.x;
  int k = (LEVEL == 1) ? (int)((float)counts[idx] * kFrac) : kcur[idx];
  if (k <= 0) {                           // wave-uniform branch (EXEC stays full)
    if (LEVEL == 1 && lane == 0) {
      kcur[idx] = -1; path[idx] = 0u; thr[idx] = 0xFFFFFFFFu;
    }
    return;
  }
  const int* h = hist + idx * 256;
  const int m  = lane & 15;               // A-matrix row / B-matrix col
  const int kh = (lane < 16) ? 0 : 2;     // K-slot half per lane group

  v8f acc = {0.f, 0.f, 0.f, 0.f, 0.f, 0.f, 0.f, 0.f};
#pragma unroll
  for (int j = 0; j < 4; ++j) {
    const int kg = 4 * j + kh;
    v2f a, bm;
    a.x  = (m >= kg)     ? 1.0f : 0.0f;   // L[m][kg]
    a.y  = (m >= kg + 1) ? 1.0f : 0.0f;   // L[m][kg+1]
    // X[k][col] = seq[col*16+k] = hist[255 - (col*16+k)]  (descending order)
    bm.x = (float)h[255 - (m * 16 + kg)];
    bm.y = (float)h[255 - (m * 16 + kg + 1)];
    acc = __builtin_amdgcn_wmma_f32_16x16x4_f32(false, a, false, bm, (short)0,
                                                acc, false, false);
  }
  // Column (chunk) totals live in acc[7], lanes 16..31. Exclusive chunk prefix:
  const int n = m;
  float O = 0.f;
#pragma unroll
  for (int cc = 0; cc < 15; ++cc) {
    float tv = __shfl(acc[7], 16 + cc, 32);
    if (cc < n) O += tv;
  }
  // Find smallest descending-seq index with cumsum >= k  (k-th largest elem).
  const float kf = (float)k;
  int best = 0x7FFFFFFF;
#pragma unroll
  for (int r = 0; r < 8; ++r) {
    int mm = (lane < 16) ? r : (r + 8);
    if (acc[r] + O >= kf) {
      int i = n * 16 + mm;
      best = best < i ? best : i;
    }
  }
#pragma unroll
  for (int off = 16; off > 0; off >>= 1) {
    int o = __shfl_xor(best, off, 32);
    best = best < o ? best : o;
  }
  const int istar = best;
  const int bin   = 255 - istar;
  // cumulative count at istar (to get remaining k for next level)
  const int nst  = istar >> 4;
  const int mst  = istar & 15;
  const int lown = (mst < 8) ? nst : (16 + nst);
  const int rown = mst & 7;
  float Ost = 0.f;
#pragma unroll
  for (int cc = 0; cc < 15; ++cc) {
    float tv = __shfl(acc[7], 16 + cc, 32);
    if (cc < nst) Ost += tv;
  }
  float cumv = 0.f;
#pragma unroll
  for (int r = 0; r < 8; ++r)
    if (r == rown) cumv = __shfl(acc[r], lown, 32);
  cumv += Ost;
  const int knext = k - ((int)(cumv + 0.5f) - h[bin]);
  if (lane == 0) {
    if (LEVEL == 1)      { path[idx]  = (unsigned)bin << 23; kcur[idx] = knext; }
    else if (LEVEL == 2) { path[idx] |= (unsigned)bin << 15; kcur[idx] = knext; }
    else if (LEVEL == 3) { path[idx] |= (unsigned)bin << 7;  kcur[idx] = knext; }
    else                 { thr[idx]   = path[idx] | (unsigned)bin; }
  }
}

// Refinement histogram: only pixels whose high bits match the selected path.
__global__ __launch_bounds__(256) void histpass_kernel(
    const float* __restrict__ prob, const unsigned char* __restrict__ lab,
    const unsigned* __restrict__ path, int* __restrict__ hist,
    int s_prev, int s_this, int binmask) {
  __shared__ unsigned spath[kBC];
  for (int i = threadIdx.x; i < kBC; i += blockDim.x) spath[i] = path[i];
  __syncthreads();
  long t = (long)blockIdx.x * blockDim.x + threadIdx.x;
  long g = t * 4;
  int  b = (int)(g >> 19);
  float4 p4 = *(const float4*)(prob + g);
  uchar4 l4 = *(const uchar4*)(lab + g);
  float pv[4] = {p4.x, p4.y, p4.z, p4.w};
  unsigned char lv[4] = {l4.x, l4.y, l4.z, l4.w};
#pragma unroll
  for (int j = 0; j < 4; ++j) {
    unsigned v = __float_as_uint(pv[j]);
    int id = b * kC + (int)lv[j];
    if ((v >> s_prev) == (spath[id] >> s_prev))
      atomicAdd(&hist[id * 256 + (int)((v >> s_this) & (unsigned)binmask)], 1);
  }
}

// Final pass: mask = p > 0.9 | bits(p) >= T[b,lab]; NLL = -log(max_prob).
__global__ __launch_bounds__(256) void final_kernel(
    const float* __restrict__ prob, const unsigned char* __restrict__ lab,
    const unsigned* __restrict__ thr, float* __restrict__ sumAcc,
    int* __restrict__ cntAcc) {
  __shared__ unsigned sthr[kBC];
  __shared__ float ssum;
  __shared__ int scnt;
  for (int i = threadIdx.x; i < kBC; i += blockDim.x) sthr[i] = thr[i];
  if (threadIdx.x == 0) { ssum = 0.f; scnt = 0; }
  __syncthreads();
  long t = (long)blockIdx.x * blockDim.x + threadIdx.x;
  long g = t * 4;
  int  b = (int)(g >> 19);
  float4 p4 = *(const float4*)(prob + g);
  uchar4 l4 = *(const uchar4*)(lab + g);
  float pv[4] = {p4.x, p4.y, p4.z, p4.w};
  unsigned char lv[4] = {l4.x, l4.y, l4.z, l4.w};
  float fs = 0.f; int ic = 0;
#pragma unroll
  for (int j = 0; j < 4; ++j) {
    unsigned v = __float_as_uint(pv[j]);
    unsigned T = sthr[b * kC + (int)lv[j]];
    if ((pv[j] > kConf) || (v >= T)) { fs -= __logf(pv[j]); ic += 1; }
  }
#pragma unroll
  for (int off = 16; off > 0; off >>= 1) {
    fs += __shfl_xor(fs, off, 32);
    ic += __shfl_xor(ic, off, 32);
  }
  if ((threadIdx.x & 31) == 0) { atomicAdd(&ssum, fs); atomicAdd(&scnt, ic); }
  __syncthreads();
  if (threadIdx.x == 0) { atomicAdd(sumAcc, ssum); atomicAdd(cntAcc, scnt); }
}

__global__ void finalize_kernel(const float* __restrict__ sumAcc,
                                const int* __restrict__ cntAcc,
                                float* __restrict__ out) {
  int c = *cntAcc;
  out[0] = (*sumAcc) / (float)(c > 0 ? c : 1);   // LAMBDA = 1
}

extern "C" void kernel_launch(void* const* d_in, const int* in_sizes, int n_in,
                              void* d_out, int out_size, void* d_ws, size_t ws_size,
                              hipStream_t stream) {
  (void)in_sizes; (void)n_in; (void)out_size; (void)ws_size;
  const float* pred = (const float*)d_in[0];
  char* ws = (char*)d_ws;
  float*         sumAcc = (float*)(ws + OFF_SUM);
  int*           cntAcc = (int*)(ws + OFF_CNT);
  int*           counts = (int*)(ws + OFF_COUNTS);
  int*           kcur   = (int*)(ws + OFF_KCUR);
  unsigned*      path   = (unsigned*)(ws + OFF_PATH);
  unsigned*      thr    = (unsigned*)(ws + OFF_THR);
  int*           h1     = (int*)(ws + OFF_H1);
  int*           h2     = (int*)(ws + OFF_H2);
  int*           h3     = (int*)(ws + OFF_H3);
  int*           h4     = (int*)(ws + OFF_H4);
  float*         prob   = (float*)(ws + OFF_PROB);
  unsigned char* lab    = (unsigned char*)(ws + OFF_LAB);

  const int nStream = (int)(kP / 1024);   // 4096 blocks, 256 thr, 4 px/thr

  zero_ws_kernel<<<(ZERO_INTS + 255) / 256, 256, 0, stream>>>((int*)ws, ZERO_INTS);
  pass1_kernel<<<nStream, 256, 0, stream>>>(pred, prob, lab, counts, h1);
  select_kernel<1><<<kBC, 32, 0, stream>>>(h1, counts, kcur, path, thr);
  histpass_kernel<<<nStream, 256, 0, stream>>>(prob, lab, path, h2, 23, 15, 255);
  select_kernel<2><<<kBC, 32, 0, stream>>>(h2, counts, kcur, path, thr);
  histpass_kernel<<<nStream, 256, 0, stream>>>(prob, lab, path, h3, 15, 7, 255);
  select_kernel<3><<<kBC, 32, 0, stream>>>(h3, counts, kcur, path, thr);
  histpass_kernel<<<nStream, 256, 0, stream>>>(prob, lab, path, h4, 7, 0, 127);
  select_kernel<4><<<kBC, 32, 0, stream>>>(h4, counts, kcur, path, thr);
  final_kernel<<<nStream, 256, 0, stream>>>(prob, lab, thr, sumAcc, cntAcc);
  finalize_kernel<<<1, 1, 0, stream>>>(sumAcc, cntAcc, (float*)d_out);
}